// LocalGrouper_65309272703070
// MI455X (gfx1250) — compile-verified
//
#include <hip/hip_runtime.h>
#include <math.h>

#define BATCH   16
#define NPTS    4096
#define GROUPS_ 1024
#define KNN     24
#define FEAT    64

typedef __attribute__((ext_vector_type(2))) float v2f;
typedef __attribute__((ext_vector_type(8))) float v8f;

#if defined(__gfx1250__) && __has_builtin(__builtin_amdgcn_wmma_f32_16x16x4_f32)
#define USE_WMMA 1
#else
#define USE_WMMA 0
#endif

// 64-bit butterfly shuffle built from two 32-bit lane shuffles (wave32).
__device__ __forceinline__ unsigned long long shflxor64(unsigned long long v, int m) {
  unsigned lo = (unsigned)(v & 0xFFFFFFFFull);
  unsigned hi = (unsigned)(v >> 32);
  lo = __shfl_xor(lo, m, 32);
  hi = __shfl_xor(hi, m, 32);
  return ((unsigned long long)hi << 32) | lo;
}

// Monotonic float -> uint map (ascending order preserved, handles negatives).
__device__ __forceinline__ unsigned fkey(float f) {
  unsigned b = __float_as_uint(f);
  return (b & 0x80000000u) ? ~b : (b | 0x80000000u);
}

// ---------------------------------------------------------------------------
// Kernel 1: furthest point sampling. One workgroup per batch, xyz in LDS,
// min_d in registers, argmax via wave32 shuffles + cross-wave LDS reduce.
// Matches reference: idx[0] = 0; step s records incoming last_idx.
// ---------------------------------------------------------------------------
__global__ __launch_bounds__(1024) void fps_kernel(const float* __restrict__ xyz,
                                                   int* __restrict__ fps) {
  const int b = blockIdx.x;
  const int tid = threadIdx.x;
  __shared__ float sx[NPTS], sy[NPTS], sz[NPTS];
  __shared__ unsigned long long wbest[32];
  __shared__ int s_last;

  const float* base = xyz + (size_t)b * NPTS * 3;
  for (int p = tid; p < NPTS; p += 1024) {
    sx[p] = base[3 * p + 0];
    sy[p] = base[3 * p + 1];
    sz[p] = base[3 * p + 2];
  }
  __syncthreads();

  float md[4] = {1e10f, 1e10f, 1e10f, 1e10f};
  int last = 0;
  for (int s = 0; s < GROUPS_; ++s) {
    if (tid == 0) fps[(b << 10) + s] = last;
    const float lx = sx[last], ly = sy[last], lz = sz[last];
    unsigned long long best = 0ull;
#pragma unroll
    for (int j = 0; j < 4; ++j) {
      const int p = tid + j * 1024;
      const float dx = sx[p] - lx, dy = sy[p] - ly, dz = sz[p] - lz;
      const float d = dx * dx + dy * dy + dz * dz;
      const float mn = fminf(md[j], d);
      md[j] = mn;
      // max-key: larger dist wins, ties -> smaller index (stable argmax).
      const unsigned long long k =
          ((unsigned long long)__float_as_uint(mn) << 12) | (unsigned)(4095 - p);
      best = (k > best) ? k : best;
    }
#pragma unroll
    for (int msk = 16; msk > 0; msk >>= 1) {
      const unsigned long long o = shflxor64(best, msk);
      best = (o > best) ? o : best;
    }
    if ((tid & 31) == 0) wbest[tid >> 5] = best;
    __syncthreads();
    if (tid < 32) {
      unsigned long long v = wbest[tid];
#pragma unroll
      for (int msk = 16; msk > 0; msk >>= 1) {
        const unsigned long long o = shflxor64(v, msk);
        v = (o > v) ? o : v;
      }
      if (tid == 0) s_last = 4095 - (int)(v & 0xFFFull);
    }
    __syncthreads();
    last = s_last;
  }
}

// ---------------------------------------------------------------------------
// Kernel 2: gather sampled anchor coordinates (first output region).
// ---------------------------------------------------------------------------
__global__ void gather_newxyz_kernel(const float* __restrict__ xyz,
                                     const int* __restrict__ fps,
                                     float* __restrict__ newxyz) {
  const int i = blockIdx.x * blockDim.x + threadIdx.x;
  if (i >= BATCH * GROUPS_) return;
  const int b = i >> 10;
  const int idx = fps[i];
  const float* s = xyz + ((size_t)b * NPTS + idx) * 3;
  float* d = newxyz + (size_t)i * 3;
  d[0] = s[0]; d[1] = s[1]; d[2] = s[2];
}

// ---------------------------------------------------------------------------
// Kernel 3: KNN via V_WMMA_F32_16X16X4_F32 distance tiles + wave-synchronous
// sorted top-24 selection. One WG = (batch, 16 anchor rows); 16 waves.
//   s(g,n) = |n|^2 - 2 g.n  (row-constant |g|^2 dropped: order-invariant)
//   A row m = (-2gx, -2gy, -2gz, 1)   B col n = (nx, ny, nz, |n|^2)
// f32 fragment layout (ISA 7.12.2): lanes 0-15 hold K={0,1}, lanes 16-31 K={2,3}.
// Tile loop is fully unrolled with a uniform wave id so no EXEC masking ever
// wraps the WMMAs (ISA 7.12: EXEC must be all 1s for WMMA).
// ---------------------------------------------------------------------------
__global__ __launch_bounds__(512) void knn_kernel(const float* __restrict__ xyz,
                                                  const float* __restrict__ newxyz,
                                                  int* __restrict__ knn) {
  const int blk = blockIdx.x;        // 0..1023
  const int b = blk >> 6;
  const int g0 = (blk & 63) << 4;
  const int tid = threadIdx.x;
  const int lane = tid & 31;
  // Wave id is uniform across the wave: tell the compiler so tile/selection
  // control flow stays scalar (no divergent exec-masked loops around WMMA).
  const int w = __builtin_amdgcn_readfirstlane(tid >> 5);   // 0..15 == row id
  const bool hic = lane >= 16;
  const int m = lane & 15;

  __shared__ float cx[1024], cy[1024], cz[1024], cw[1024];  // chunk coords + |n|^2
  __shared__ float dist[16][1024];                          // 64KB distance tile
  __shared__ unsigned long long cand[16][4 * KNN];          // per-chunk top-24 keys
  __shared__ float ga[16][3];                               // anchor coords

  if (tid < 16) {
    const float* p = newxyz + ((size_t)b * GROUPS_ + g0 + tid) * 3;
    ga[tid][0] = p[0]; ga[tid][1] = p[1]; ga[tid][2] = p[2];
  }
  __syncthreads();

#if USE_WMMA
  v2f afrag;
  afrag.x = hic ? (-2.0f * ga[m][2]) : (-2.0f * ga[m][0]);
  afrag.y = hic ? 1.0f : (-2.0f * ga[m][1]);
#endif

  const float* xb = xyz + (size_t)b * NPTS * 3;

  for (int c = 0; c < 4; ++c) {
    const int nbase = c << 10;
    for (int i = tid; i < 1024; i += 512) {
      const float* p = xb + (size_t)(nbase + i) * 3;
      const float x = p[0], y = p[1], z = p[2];
      cx[i] = x; cy[i] = y; cz[i] = z; cw[i] = x * x + y * y + z * z;
    }
    __syncthreads();

    // Prefetch next chunk's coordinates into cache while we compute
    // (lowers to global_prefetch_b8: no counters, pure latency overlap).
    if (c < 3) {
      const int pn = ((c + 1) << 10) + tid * 2;
      __builtin_prefetch(xb + (size_t)pn * 3, 0, 0);
    }

#if USE_WMMA
#pragma unroll
    for (int tt = 0; tt < 4; ++tt) {             // 4 WMMA tiles per wave
      const int nl = ((w + tt * 16) << 4) + m;
      v2f bfrag;
      bfrag.x = hic ? cz[nl] : cx[nl];
      bfrag.y = hic ? cw[nl] : cy[nl];
      v8f acc = {};
      acc = __builtin_amdgcn_wmma_f32_16x16x4_f32(false, afrag, false, bfrag,
                                                  (short)0, acc, false, false);
#pragma unroll
      for (int r = 0; r < 8; ++r) {
        dist[r + (hic ? 8 : 0)][nl] = acc[r];    // D: row = vgpr + 8*(lane>=16)
      }
    }
#else
    for (int e = tid; e < 16 * 1024; e += 512) {
      const int row = e >> 10, col = e & 1023;
      dist[row][col] = cw[col] - 2.0f * (ga[row][0] * cx[col] +
                                         ga[row][1] * cy[col] +
                                         ga[row][2] * cz[col]);
    }
#endif
    __syncthreads();

    // Per-chunk sorted top-24: repeated lexicographic (dist, idx) argmin.
    unsigned long long prev = 0ull;
    for (int sel = 0; sel < KNN; ++sel) {
      unsigned long long best = ~0ull;
      for (int i = lane; i < 1024; i += 32) {
        const unsigned long long key =
            ((unsigned long long)fkey(dist[w][i]) << 12) | (unsigned)(nbase + i);
        if (key > prev && key < best) best = key;
      }
#pragma unroll
      for (int msk = 16; msk > 0; msk >>= 1) {
        const unsigned long long o = shflxor64(best, msk);
        best = (o < best) ? o : best;
      }
      prev = best;
      if (lane == 0) cand[w][c * KNN + sel] = best;
    }
    __syncthreads();
  }

  // Merge 4x24 candidates -> global sorted top-24, emit neighbor indices.
  unsigned long long prev = 0ull;
  int* out = knn + ((size_t)b * GROUPS_ + g0 + w) * KNN;
  for (int sel = 0; sel < KNN; ++sel) {
    unsigned long long best = ~0ull;
    for (int i = lane; i < 4 * KNN; i += 32) {
      const unsigned long long key = cand[w][i];
      if (key > prev && key < best) best = key;
    }
#pragma unroll
    for (int msk = 16; msk > 0; msk >>= 1) {
      const unsigned long long o = shflxor64(best, msk);
      best = (o < best) ? o : best;
    }
    prev = best;
    if (lane == 0) out[sel] = (int)(best & 0xFFFull);
  }
}

// ---------------------------------------------------------------------------
// Kernel 4: per-group augmented-xyz normalization + grouped feature assembly.
// One workgroup per (b, g).
// ---------------------------------------------------------------------------
__global__ __launch_bounds__(256) void finalize_kernel(
    const float* __restrict__ xyz, const float* __restrict__ points,
    const int* __restrict__ fps, const int* __restrict__ knn,
    const float* __restrict__ newxyz,
    float* __restrict__ out_aug, float* __restrict__ out_grp) {
  const int bg = blockIdx.x;   // b*1024 + g
  const int b = bg >> 10;
  const int tid = threadIdx.x;

  __shared__ float aug[KNN][10];
  __shared__ float mean_s[10], inv_s[10];
  __shared__ int sidx[KNN];
  __shared__ int sf;
  __shared__ float anc[3];

  if (tid < KNN) sidx[tid] = knn[(size_t)bg * KNN + tid];
  if (tid == KNN) sf = fps[bg];
  if (tid < 3) anc[tid] = newxyz[(size_t)bg * 3 + tid];
  __syncthreads();

  if (tid < KNN) {
    const float* p = xyz + ((size_t)b * NPTS + sidx[tid]) * 3;
    const float gx = p[0], gy = p[1], gz = p[2];
    const float vx = gx - anc[0], vy = gy - anc[1], vz = gz - anc[2];
    const float ad = sqrtf(vx * vx + vy * vy + vz * vz);
    aug[tid][0] = ad;
    aug[tid][1] = vx; aug[tid][2] = vy; aug[tid][3] = vz;
    aug[tid][4] = anc[0]; aug[tid][5] = anc[1]; aug[tid][6] = anc[2];
    aug[tid][7] = gx; aug[tid][8] = gy; aug[tid][9] = gz;
  }
  __syncthreads();

  if (tid < 10) {
    float s = 0.0f;
    for (int k = 0; k < KNN; ++k) s += aug[k][tid];
    const float mu = s / (float)KNN;
    float v = 0.0f;
    for (int k = 0; k < KNN; ++k) {
      const float d = aug[k][tid] - mu;
      v += d * d;
    }
    v /= (float)(KNN - 1);
    mean_s[tid] = mu;
    inv_s[tid] = 1.0f / (sqrtf(v) + 1e-8f);
  }
  __syncthreads();

  if (tid < KNN * 10) {
    const int k = tid / 10, ch = tid % 10;
    out_aug[((size_t)bg * KNN + k) * 10 + ch] = (aug[k][ch] - mean_s[ch]) * inv_s[ch];
  }

  const float* pb = points + (size_t)b * NPTS * FEAT;
  for (int e = tid; e < KNN * 2 * FEAT; e += 256) {
    const int k = e >> 7, ch = e & 127;
    const float v = (ch < FEAT) ? pb[(size_t)sidx[k] * FEAT + ch]
                                : pb[(size_t)sf * FEAT + (ch - FEAT)];
    out_grp[(size_t)bg * KNN * 2 * FEAT + e] = v;
  }
}

// ---------------------------------------------------------------------------
extern "C" void kernel_launch(void* const* d_in, const int* in_sizes, int n_in,
                              void* d_out, int out_size, void* d_ws, size_t ws_size,
                              hipStream_t stream) {
  const float* xyz = (const float*)d_in[0];      // [16,4096,3]
  const float* points = (const float*)d_in[1];   // [16,4096,64]

  float* out = (float*)d_out;
  float* out_newxyz = out;                                        // 16*1024*3
  float* out_aug = out_newxyz + (size_t)BATCH * GROUPS_ * 3;      // 16*1024*24*10
  float* out_grp = out_aug + (size_t)BATCH * GROUPS_ * KNN * 10;  // 16*1024*24*128

  int* fps = (int*)d_ws;                    // 16*1024 ints
  int* knn = fps + BATCH * GROUPS_;         // 16*1024*24 ints

  fps_kernel<<<BATCH, 1024, 0, stream>>>(xyz, fps);
  gather_newxyz_kernel<<<(BATCH * GROUPS_ + 255) / 256, 256, 0, stream>>>(
      xyz, fps, out_newxyz);
  knn_kernel<<<BATCH * (GROUPS_ / 16), 512, 0, stream>>>(xyz, out_newxyz, knn);
  finalize_kernel<<<BATCH * GROUPS_, 256, 0, stream>>>(
      xyz, points, fps, knn, out_newxyz, out_aug, out_grp);
}